// AttnSum3d_42082089566246
// MI455X (gfx1250) — compile-verified
//
#include <hip/hip_runtime.h>
#include <math.h>

#define B_ 16
#define L_ 2048
#define D_ 128
#define MB 128       // query rows per workgroup
#define NB 128       // key block staged in LDS
#define STRIDE 132   // padded LDS row stride (floats): 132%64=4 kills bank conflicts, 132*4%16==0 keeps float4 alignment

typedef __attribute__((ext_vector_type(2))) float v2f;
typedef __attribute__((ext_vector_type(8))) float v8f;

// D(16x16,f32) += A(16x4,f32) * B(4x16,f32) -- CDNA5 V_WMMA_F32_16X16X4_F32
__device__ __forceinline__ v8f wmma_f32_k4(float a0, float a1, float b0, float b1, v8f c) {
  v2f a = {a0, a1};
  v2f b = {b0, b1};
  return __builtin_amdgcn_wmma_f32_16x16x4_f32(false, a, false, b, (short)0, c, false, false);
}

// ---------------- kernel 1: x = input * mask, zero the out head -------------
__global__ void k_premul(const float* __restrict__ in, const float* __restrict__ mask,
                         float* __restrict__ xm, float* __restrict__ out) {
  int i4 = blockIdx.x * blockDim.x + threadIdx.x;      // float4 index over B*L*D/4
  float4 v = ((const float4*)in)[i4];
  float mval = mask[i4 >> 5];                          // 32 float4 per (b,l) row
  v.x *= mval; v.y *= mval; v.z *= mval; v.w *= mval;
  ((float4*)xm)[i4] = v;
  if (i4 < B_ * D_) out[i4] = 0.0f;                    // zero [B,1,D] accumulators
}

// ---------------- kernel 2: flash pass -> rowmax, rowsumexp; zero rbuf ------
__global__ void __launch_bounds__(256, 1)
k_stats(const float* __restrict__ xm, float* __restrict__ mmax,
        float* __restrict__ denom, float* __restrict__ rbuf) {
  extern __shared__ float lds[];
  float* xq = lds;
  float* xk = lds + MB * STRIDE;
  const int b = blockIdx.y;
  const int row0 = blockIdx.x * MB;
  const int tid = threadIdx.x;
  const int lane = tid & 31;
  const int wave = tid >> 5;

  if (tid < MB) rbuf[b * L_ + row0 + tid] = 0.0f;      // runs before k_accum (stream order)

  // stage the 128x128 query block
  {
    const float4* src = (const float4*)(xm + ((size_t)b * L_ + row0) * D_);
#pragma unroll 4
    for (int j = 0; j < (MB * (D_ / 4)) / 256; ++j) {
      int idx = tid + j * 256;
      int row = idx >> 5, c4 = idx & 31;
      *(float4*)&xq[row * STRIDE + c4 * 4] = src[row * 32 + c4];
    }
  }
  __syncthreads();

  // hoist A fragments for this wave's 16 rows (whole K=128) into registers
  const int mloc = wave * 16 + (lane & 15);
  const int k0 = (lane >> 4) * 2;
  float af[64];
#pragma unroll
  for (int t = 0; t < 32; ++t) {
    float2 v = *(const float2*)&xq[mloc * STRIDE + 4 * t + k0];
    af[2 * t] = v.x; af[2 * t + 1] = v.y;
  }

  float mr[8], sr[8];
#pragma unroll
  for (int r = 0; r < 8; ++r) { mr[r] = -1.0e30f; sr[r] = 0.0f; }

  const int nlane = lane & 15;
  for (int kb = 0; kb < L_ / NB; ++kb) {
    __syncthreads();
    const float4* ksrc = (const float4*)(xm + ((size_t)b * L_ + kb * NB) * D_);
#pragma unroll 4
    for (int j = 0; j < (NB * (D_ / 4)) / 256; ++j) {
      int idx = tid + j * 256;
      int row = idx >> 5, c4 = idx & 31;
      *(float4*)&xk[row * STRIDE + c4 * 4] = ksrc[row * 32 + c4];
    }
    __syncthreads();

    for (int nt = 0; nt < NB / 16; ++nt) {
      v8f c = {0.f, 0.f, 0.f, 0.f, 0.f, 0.f, 0.f, 0.f};
      const float* bbase = &xk[(nt * 16 + nlane) * STRIDE + k0];
#pragma unroll
      for (int t = 0; t < 32; ++t) {
        float2 bv = *(const float2*)(bbase + 4 * t);
        c = wmma_f32_k4(af[2 * t], af[2 * t + 1], bv.x, bv.y, c);
      }
      // online softmax update: lane holds 8 rows, one column of this tile
#pragma unroll
      for (int r = 0; r < 8; ++r) {
        float v = c[r];
        float nm = fmaxf(mr[r], v);
        sr[r] = sr[r] * __expf(mr[r] - nm) + __expf(v - nm);
        mr[r] = nm;
      }
    }
  }

  // merge (m,s) across the 16 lanes of each half-wave
#pragma unroll
  for (int off = 1; off < 16; off <<= 1) {
#pragma unroll
    for (int r = 0; r < 8; ++r) {
      float om = __shfl_xor(mr[r], off, 32);
      float os = __shfl_xor(sr[r], off, 32);
      float nm = fmaxf(mr[r], om);
      sr[r] = sr[r] * __expf(mr[r] - nm) + os * __expf(om - nm);
      mr[r] = nm;
    }
  }
  if ((lane & 15) == 0) {
    int rb = row0 + wave * 16 + (lane >> 4) * 8;
#pragma unroll
    for (int r = 0; r < 8; ++r) {
      mmax[b * L_ + rb + r] = mr[r];
      denom[b * L_ + rb + r] = sr[r];
    }
  }
}

// ---------------- kernel 3: recompute P, accumulate column sums r[l] --------
__global__ void __launch_bounds__(256, 1)
k_accum(const float* __restrict__ xm, const float* __restrict__ mmax,
        const float* __restrict__ denom, float* __restrict__ rbuf) {
  extern __shared__ float lds[];
  float* xq = lds;
  float* xk = lds + MB * STRIDE;
  __shared__ float ldsr[NB];
  const int b = blockIdx.y;
  const int row0 = blockIdx.x * MB;
  const int tid = threadIdx.x;
  const int lane = tid & 31;
  const int wave = tid >> 5;

  {
    const float4* src = (const float4*)(xm + ((size_t)b * L_ + row0) * D_);
#pragma unroll 4
    for (int j = 0; j < (MB * (D_ / 4)) / 256; ++j) {
      int idx = tid + j * 256;
      int row = idx >> 5, c4 = idx & 31;
      *(float4*)&xq[row * STRIDE + c4 * 4] = src[row * 32 + c4];
    }
  }
  __syncthreads();

  const int mloc = wave * 16 + (lane & 15);
  const int k0 = (lane >> 4) * 2;
  float af[64];
#pragma unroll
  for (int t = 0; t < 32; ++t) {
    float2 v = *(const float2*)&xq[mloc * STRIDE + 4 * t + k0];
    af[2 * t] = v.x; af[2 * t + 1] = v.y;
  }

  // per-lane softmax stats for its 8 rows
  const int rb = row0 + wave * 16 + (lane >> 4) * 8;
  float ms[8], idn[8];
#pragma unroll
  for (int r = 0; r < 8; ++r) {
    ms[r] = mmax[b * L_ + rb + r];
    idn[r] = 1.0f / denom[b * L_ + rb + r];
  }

  const int nlane = lane & 15;
  for (int kb = 0; kb < L_ / NB; ++kb) {
    __syncthreads();
    const float4* ksrc = (const float4*)(xm + ((size_t)b * L_ + kb * NB) * D_);
#pragma unroll 4
    for (int j = 0; j < (NB * (D_ / 4)) / 256; ++j) {
      int idx = tid + j * 256;
      int row = idx >> 5, c4 = idx & 31;
      *(float4*)&xk[row * STRIDE + c4 * 4] = ksrc[row * 32 + c4];
    }
    if (tid < NB) ldsr[tid] = 0.0f;
    __syncthreads();

    for (int nt = 0; nt < NB / 16; ++nt) {
      v8f c = {0.f, 0.f, 0.f, 0.f, 0.f, 0.f, 0.f, 0.f};
      const float* bbase = &xk[(nt * 16 + nlane) * STRIDE + k0];
#pragma unroll
      for (int t = 0; t < 32; ++t) {
        float2 bv = *(const float2*)(bbase + 4 * t);
        c = wmma_f32_k4(af[2 * t], af[2 * t + 1], bv.x, bv.y, c);
      }
      float colp = 0.0f;
#pragma unroll
      for (int r = 0; r < 8; ++r) colp += __expf(c[r] - ms[r]) * idn[r];
      colp += __shfl_xor(colp, 16, 32);                 // rows 0-7 + rows 8-15
      if (lane < 16) atomicAdd(&ldsr[nt * 16 + lane], colp);
    }
    __syncthreads();
    if (tid < NB) atomicAdd(&rbuf[b * L_ + kb * NB + tid], ldsr[tid]);
  }
}

// ---------------- kernel 4: out = (1/L) * r . X ; attn_mean = 1/L -----------
__global__ void k_final(const float* __restrict__ xm, const float* __restrict__ rbuf,
                        float* __restrict__ out) {
  const int b = blockIdx.y;
  const int chunk = blockIdx.x;                        // 0..7
  const int tid = threadIdx.x;                         // 0..127 = d
  const int l0 = chunk * (L_ / 8);
  float acc = 0.0f;
  for (int l = l0; l < l0 + L_ / 8; ++l)
    acc += rbuf[b * L_ + l] * xm[((size_t)b * L_ + l) * D_ + tid];
  atomicAdd(&out[b * D_ + tid], acc * (1.0f / (float)L_));

  float* attn = out + B_ * D_;                         // [B,L] second output
#pragma unroll
  for (int i = 0; i < 2; ++i)
    attn[b * L_ + l0 + i * 128 + tid] = 1.0f / (float)L_;   // exact: softmax rows sum to 1
}

extern "C" void kernel_launch(void* const* d_in, const int* in_sizes, int n_in,
                              void* d_out, int out_size, void* d_ws, size_t ws_size,
                              hipStream_t stream) {
  const float* in = (const float*)d_in[0];
  const float* mask = (const float*)d_in[1];
  float* out = (float*)d_out;

  float* xm = (float*)d_ws;                            // 16 MB
  float* mmax = xm + (size_t)B_ * L_ * D_;             // 128 KB
  float* denom = mmax + (size_t)B_ * L_;               // 128 KB
  float* rbuf = denom + (size_t)B_ * L_;               // 128 KB

  const size_t ldsBytes = (size_t)2 * MB * STRIDE * sizeof(float);  // 132 KB

  k_premul<<<dim3((B_ * L_ * D_ / 4) / 256), dim3(256), 0, stream>>>(in, mask, xm, out);
  k_stats<<<dim3(L_ / MB, B_), dim3(256), ldsBytes, stream>>>(xm, mmax, denom, rbuf);
  k_accum<<<dim3(L_ / MB, B_), dim3(256), ldsBytes, stream>>>(xm, mmax, denom, rbuf);
  k_final<<<dim3(8, B_), dim3(128), 0, stream>>>(xm, rbuf, out);
}